// ICTDIrrepsE3Conv_29394756174147
// MI455X (gfx1250) — compile-verified
//
#include <hip/hip_runtime.h>
#include <hip/hip_bf16.h>

// ---------------------------------------------------------------------------
// E(3) conv: only l2==0 paths (p = 0,3,9) contribute; CG is identity there.
//   feat[e] = gate_p * (1/8) * Y_lo[o] * S_p[w],
//   S_p[w]  = sum_{u,v} Ai[src,u]*Ai[dst,v]*tp_w[p,u,v,w]
// node_out = segment_mean over edge_dst.
// WMMA: V_WMMA_F32_16X16X4_F32 tiles of 16 edges for the radial MLP + TP.
// ---------------------------------------------------------------------------

typedef float v2f __attribute__((ext_vector_type(2)));
typedef float v8f __attribute__((ext_vector_type(8)));

#define MULC     8
#define MULOUT   16
#define NBAS     16
#define NPATH    15
#define FEAT     144          // 16*1 + 16*3 + 16*5
#define S3f      1.7320508075688772f
#define S5f      2.2360679774997896f
#define S15f     3.8729833462074170f
#define RINV     (17.0f / 5.0f)
#define ESCALE   (4.0f / 1.12f)
#define PNORM    0.125f       // 1/sqrt(8*8)

// ---- LDS layout (floats) ---------------------------------------------------
#define SW1_N   (16 * 80)
#define SW2_N   (64 * 80)
#define SW3_N   (64 * 48)
#define STP_N   (64 * 80)
#define WGT_N   (SW1_N + SW2_N + SW3_N + STP_N + 64 + 64 + 16)
// per-wave: emb 32x20, H0 16x68, H1 16x68, gates 32x20, outer 16x68, S 32x52
#define PW_EMB  (32 * 20)
#define PW_H    (16 * 68)
#define PW_GAT  (32 * 20)
#define PW_OUT  (16 * 68)
#define PW_S    (32 * 52)
#define PW_N    (PW_EMB + 2 * PW_H + PW_GAT + PW_OUT + PW_S)
#define NWAVES  4
#define SMEM_FLOATS (WGT_N + NWAVES * PW_N)

// fast silu: x * rcp(1 + exp(-x)) -- v_rcp_f32 (1 ulp) instead of IEEE divide
__device__ __forceinline__ float fast_silu(float t) {
  return t * __builtin_amdgcn_rcpf(1.0f + __expf(-t));
}

// ---- WMMA fragment helpers (f32 16x16x4, ISA 7.12.2 layouts) --------------
// A 16x4: lane&15 = M row; lanes>=16 hold K+2. VGPR j -> K = kh + j.
// B 4x16: lane&15 = N col; VGPR j -> K = kh + j (mirror of A).
// C/D:    VGPR j -> M = j + 8*(lane>=16), N = lane&15.

template <int K>
__device__ __forceinline__ v8f wmma_gemm(const float* aBuf, int Sa,
                                         const float* wBuf, int Sb,
                                         int nbase, int lane) {
  const int idx = lane & 15;
  const int kh  = (lane >> 4) << 1;
  const float* ap = aBuf + idx * Sa + kh;
  const float* bp = wBuf + kh * Sb + nbase + idx;
  v8f acc = {};
#pragma unroll
  for (int k = 0; k < K; k += 4) {
    v2f a = {ap[k], ap[k + 1]};
    v2f b = {bp[k * Sb], bp[k * Sb + Sb]};
    acc = __builtin_amdgcn_wmma_f32_16x16x4_f32(false, a, false, b,
                                                (short)0, acc, false, false);
  }
  return acc;
}

__device__ __forceinline__ void stD(float* buf, int S, int nbase, int lane, v8f d) {
  const int col = nbase + (lane & 15);
  float* p = buf + ((lane >> 4) << 3) * S + col;
#pragma unroll
  for (int j = 0; j < 8; ++j) p[j * S] = d[j];
}

__device__ __forceinline__ v8f bias_silu(v8f x, float b) {
#pragma unroll
  for (int j = 0; j < 8; ++j) x[j] = fast_silu(x[j] + b);
  return x;
}

// ---------------------------------------------------------------------------
__global__ __launch_bounds__(256) void zero_kernel(float* out, int n_out,
                                                   float* counts, int n_cnt) {
  int i = blockIdx.x * blockDim.x + threadIdx.x;
  if (i < n_out) out[i] = 0.0f;
  if (i < n_cnt) counts[i] = 0.0f;
}

__global__ __launch_bounds__(128) void node_ai_kernel(
    const int* __restrict__ A, const float* __restrict__ embt,
    const float* __restrict__ w1, const float* __restrict__ b1,
    const float* __restrict__ w2, const float* __restrict__ b2,
    float* __restrict__ Ai, int N) {
  int n = blockIdx.x * blockDim.x + threadIdx.x;
  if (n >= N) return;
  int a = A[n];
  float e[16];
#pragma unroll
  for (int k = 0; k < 16; ++k) e[k] = embt[a * 16 + k];
  float acc[8];
#pragma unroll
  for (int v = 0; v < 8; ++v) acc[v] = b2[v];
  for (int j = 0; j < 64; ++j) {
    float h = b1[j];
#pragma unroll
    for (int k = 0; k < 16; ++k) h += e[k] * w1[k * 64 + j];
    h = fast_silu(h);
#pragma unroll
    for (int v = 0; v < 8; ++v) acc[v] += h * w2[j * 8 + v];
  }
#pragma unroll
  for (int v = 0; v < 8; ++v) Ai[n * 8 + v] = acc[v];
}

__global__ __launch_bounds__(128) void edge_kernel(
    const float* __restrict__ pos, const int* __restrict__ batch,
    const int* __restrict__ esrc, const int* __restrict__ edst,
    const float* __restrict__ shifts, const float* __restrict__ cell,
    const float* __restrict__ fw1, const float* __restrict__ fb1,
    const float* __restrict__ fw2, const float* __restrict__ fb2,
    const float* __restrict__ fw3, const float* __restrict__ fb3,
    const float* __restrict__ tpw, const float* __restrict__ Ai,
    float* __restrict__ out, float* __restrict__ counts, int E) {
  extern __shared__ float sm[];
  const int tid = threadIdx.x;
  const int lane = tid & 31;
  const int wv = tid >> 5;
  const int idx = lane & 15;

  float* sW1 = sm;
  float* sW2 = sW1 + SW1_N;
  float* sW3 = sW2 + SW2_N;
  float* sTP = sW3 + SW3_N;
  float* sB1 = sTP + STP_N;
  float* sB2 = sB1 + 64;
  float* sB3 = sB2 + 64;
  float* wb   = sB3 + 16 + wv * PW_N;
  float* sEmb = wb;                   // 32 x 20
  float* sH0  = sEmb + PW_EMB;        // 16 x 68
  float* sH1  = sH0 + PW_H;           // 16 x 68
  float* sGat = sH1 + PW_H;           // 32 x 20
  float* sOut = sGat + PW_GAT;        // 16 x 68
  float* sS   = sOut + PW_OUT;        // 32 x 52

  // ---- stage weights into LDS (cooperative, padded strides) ----
  for (int i = tid; i < 16 * 64; i += 128) sW1[(i >> 6) * 80 + (i & 63)] = fw1[i];
  for (int i = tid; i < 64 * 64; i += 128) sW2[(i >> 6) * 80 + (i & 63)] = fw2[i];
  for (int i = tid; i < 64 * 48; i += 128) {
    int r = i / 48, c = i % 48;
    sW3[r * 48 + c] = (c < NPATH) ? fw3[r * NPATH + c] : 0.0f;
  }
  for (int i = tid; i < 64 * 48; i += 128) {
    int uv = i / 48, r = i % 48;
    int pi = r >> 4, w = r & 15;
    int p = (pi == 0) ? 0 : ((pi == 1) ? 3 : 9);
    sTP[uv * 80 + r] = tpw[(p * 64 + uv) * 16 + w];
  }
  if (tid < 64) { sB1[tid] = fb1[tid]; sB2[tid] = fb2[tid]; }
  if (tid < 16) sB3[tid] = (tid < NPATH) ? fb3[tid] : 0.0f;
  __syncthreads();

  // ---- scalar per-edge phase (lane owns one edge) ----
  long ge = (long)blockIdx.x * 128 + tid;
  bool valid = ge < (long)E;
  int e = valid ? (int)ge : (E - 1);
  int src = esrc[e], dst = edst[e];

  float shx = shifts[e * 3 + 0], shy = shifts[e * 3 + 1], shz = shifts[e * 3 + 2];
  const float* C = cell + batch[src] * 9;
  float vx = pos[dst * 3 + 0] - pos[src * 3 + 0] + shx * C[0] + shy * C[3] + shz * C[6];
  float vy = pos[dst * 3 + 1] - pos[src * 3 + 1] + shx * C[1] + shy * C[4] + shz * C[7];
  float vz = pos[dst * 3 + 2] - pos[src * 3 + 2] + shx * C[2] + shy * C[5] + shz * C[8];
  float len = sqrtf(vx * vx + vy * vy + vz * vz);
  float rl = 1.0f / fmaxf(len, 1e-8f);
  float nx = vx * rl, ny = vy * rl, nz = vz * rl;

  float Y1v[3] = {S3f * ny, S3f * nz, S3f * nx};
  float Y2v[5] = {S15f * nx * ny, S15f * ny * nz, 0.5f * S5f * (3.0f * nz * nz - 1.0f),
                  S15f * nx * nz, 0.5f * S15f * (nx * nx - ny * ny)};

  {  // radial Gaussian basis -> LDS (A rows, stride 20)
    float* row = sEmb + lane * 20;
    float t = len * RINV;
#pragma unroll
    for (int k = 0; k < NBAS; ++k) {
      float d = t - (float)(k + 1);
      row[k] = __expf(-d * d) * ESCALE;
    }
  }

  float f1[8], x2[8];
#pragma unroll
  for (int u = 0; u < 8; ++u) { f1[u] = Ai[src * 8 + u]; x2[u] = Ai[dst * 8 + u]; }
  if (valid) atomicAdd(&counts[dst], 1.0f);

  // ---- two 16-edge WMMA tiles per wave ----
#pragma unroll
  for (int t = 0; t < 2; ++t) {
    const float* aEmb = sEmb + t * 16 * 20;

    // H0 = silu(X @ W1 + b1) : 16x64
#pragma unroll
    for (int nt = 0; nt < 4; ++nt) {
      v8f acc = wmma_gemm<16>(aEmb, 20, sW1, 80, nt * 16, lane);
      acc = bias_silu(acc, sB1[nt * 16 + idx]);
      stD(sH0, 68, nt * 16, lane, acc);
    }
    // H1 = silu(H0 @ W2 + b2) : 16x64
#pragma unroll
    for (int nt = 0; nt < 4; ++nt) {
      v8f acc = wmma_gemm<64>(sH0, 68, sW2, 80, nt * 16, lane);
      acc = bias_silu(acc, sB2[nt * 16 + idx]);
      stD(sH1, 68, nt * 16, lane, acc);
    }
    // gates = H1 @ W3 + b3 : 16x16 (cols = paths, padded)
    {
      v8f acc = wmma_gemm<64>(sH1, 68, sW3, 48, 0, lane);
      float b = sB3[idx];
#pragma unroll
      for (int j = 0; j < 8; ++j) acc[j] += b;
      stD(sGat + t * 16 * 20, 20, 0, lane, acc);
    }
    // outer rows for this tile: outer[e][u*8+v] = f1[u]*x2[v]
    if ((lane >> 4) == t) {
      float* row = sOut + idx * 68;
#pragma unroll
      for (int u = 0; u < 8; ++u)
#pragma unroll
        for (int v = 0; v < 8; ++v) row[u * 8 + v] = f1[u] * x2[v];
    }
    // S = outer(16x64) @ tpWcat(64x48)
#pragma unroll
    for (int pt = 0; pt < 3; ++pt) {
      v8f acc = wmma_gemm<64>(sOut, 68, sTP, 80, pt * 16, lane);
      stD(sS + t * 16 * 52, 52, pt * 16, lane, acc);
    }
  }

  // ---- scatter: 144 features -> node accumulator ----
  if (valid) {
    const float* gRow = sGat + lane * 20;
    const float* sRow = sS + lane * 52;
    float g0 = gRow[0] * PNORM, g3 = gRow[3] * PNORM, g9 = gRow[9] * PNORM;
    float* on = out + (size_t)dst * FEAT;
#pragma unroll
    for (int w = 0; w < 16; ++w) atomicAdd(&on[w], g0 * sRow[w]);
#pragma unroll
    for (int w = 0; w < 16; ++w) {
      float sv = g3 * sRow[16 + w];
#pragma unroll
      for (int o = 0; o < 3; ++o) atomicAdd(&on[16 + w * 3 + o], sv * Y1v[o]);
    }
#pragma unroll
    for (int w = 0; w < 16; ++w) {
      float sv = g9 * sRow[32 + w];
#pragma unroll
      for (int o = 0; o < 5; ++o) atomicAdd(&on[64 + w * 5 + o], sv * Y2v[o]);
    }
  }
}

__global__ __launch_bounds__(256) void norm_kernel(float* out,
                                                   const float* counts, int tot) {
  int i = blockIdx.x * blockDim.x + threadIdx.x;
  if (i < tot)
    out[i] = out[i] * __builtin_amdgcn_rcpf(fmaxf(counts[i / FEAT], 1.0f));
}

// ---------------------------------------------------------------------------
extern "C" void kernel_launch(void* const* d_in, const int* in_sizes, int n_in,
                              void* d_out, int out_size, void* d_ws, size_t ws_size,
                              hipStream_t stream) {
  (void)n_in; (void)out_size; (void)ws_size;
  const float* pos    = (const float*)d_in[0];
  const int*   A      = (const int*)d_in[1];
  const int*   batch  = (const int*)d_in[2];
  const int*   esrc   = (const int*)d_in[3];
  const int*   edst   = (const int*)d_in[4];
  const float* shifts = (const float*)d_in[5];
  const float* cell   = (const float*)d_in[6];
  const float* embt   = (const float*)d_in[7];
  const float* mw1    = (const float*)d_in[8];
  const float* mb1    = (const float*)d_in[9];
  const float* mw2    = (const float*)d_in[10];
  const float* mb2    = (const float*)d_in[11];
  const float* fw1    = (const float*)d_in[12];
  const float* fb1    = (const float*)d_in[13];
  const float* fw2    = (const float*)d_in[14];
  const float* fb2    = (const float*)d_in[15];
  const float* fw3    = (const float*)d_in[16];
  const float* fb3    = (const float*)d_in[17];
  const float* tpw    = (const float*)d_in[18];

  const int N = in_sizes[1];
  const int E = in_sizes[3];
  float* out = (float*)d_out;
  float* ws = (float*)d_ws;
  float* counts = ws;          // N floats
  float* Ai = ws + N;          // N*8 floats

  const int totOut = N * FEAT;
  zero_kernel<<<(totOut + 255) / 256, 256, 0, stream>>>(out, totOut, counts, N);
  node_ai_kernel<<<(N + 127) / 128, 128, 0, stream>>>(A, embt, mw1, mb1, mw2, mb2, Ai, N);
  const size_t shb = (size_t)SMEM_FLOATS * sizeof(float);
  edge_kernel<<<(E + 127) / 128, 128, shb, stream>>>(
      pos, batch, esrc, edst, shifts, cell,
      fw1, fb1, fw2, fb2, fw3, fb3, tpw, Ai, out, counts, E);
  norm_kernel<<<(totOut + 255) / 256, 256, 0, stream>>>(out, counts, totOut);
}